// CapsuleLayer_34067680592338
// MI455X (gfx1250) — compile-verified
//
#include <hip/hip_runtime.h>

// Capsule projection + capsule-sum fused as skinny GEMM:
//   Wsum[f,d] = sum_c capsules[f, c*16+d]      (256 x 16)
//   out[m,d]  = sum_f x[m,f] * Wsum[f,d]       (M=100352, K=256, N=16)
// Bandwidth-bound (~110 MB @ 23.3 TB/s); compute done with fp32 WMMA
// (v_wmma_f32_16x16x4_f32) to preserve full precision for free.

typedef __attribute__((ext_vector_type(2))) float v2f;
typedef __attribute__((ext_vector_type(8))) float v8f;

#define F_DIM 256
#define CAPS 10
#define CDIM 16
#define LDS_STRIDE 17   // pad 16 -> 17 dwords: conflict-free B-fragment reads

// Kernel 1: reduce capsules (256 x 10 x 16) -> Wsum (256 x 16) in workspace.
__global__ void caps_reduce_kernel(const float* __restrict__ caps,
                                   float* __restrict__ wsum) {
    int idx = blockIdx.x * blockDim.x + threadIdx.x;   // 0..4095
    if (idx >= F_DIM * CDIM) return;
    int f = idx >> 4;
    int d = idx & 15;
    const float* p = caps + f * (CAPS * CDIM) + d;
    float s = 0.f;
#pragma unroll
    for (int c = 0; c < CAPS; ++c) s += p[c * CDIM];
    wsum[f * CDIM + d] = s;
}

// Kernel 2: GEMM. One 16-row M-tile per wave, 8 waves per block.
// FROM_CAPS=true builds Wsum in-block from capsules (ws_size fallback).
template <bool FROM_CAPS>
__global__ __launch_bounds__(256) void caps_gemm_kernel(
    const float* __restrict__ x,
    const float* __restrict__ w,   // Wsum (256x16) or capsules (256x160)
    float* __restrict__ out,
    int numTiles) {
    __shared__ float lw[F_DIM * LDS_STRIDE];

    const int tid = threadIdx.x;

    // ---- Stage Wsum into LDS, padded stride 17: lw[k*17 + n] = Wsum[k][n]
    {
        const int k = tid;          // 256 threads == 256 K rows
        float acc[CDIM];
        if (FROM_CAPS) {
#pragma unroll
            for (int n = 0; n < CDIM; ++n) acc[n] = 0.f;
            const float* cp = w + (size_t)k * (CAPS * CDIM);
#pragma unroll
            for (int c = 0; c < CAPS; ++c) {
                const float4* q = (const float4*)(cp + c * CDIM);
#pragma unroll
                for (int j = 0; j < 4; ++j) {
                    float4 v = q[j];
                    acc[4 * j + 0] += v.x;
                    acc[4 * j + 1] += v.y;
                    acc[4 * j + 2] += v.z;
                    acc[4 * j + 3] += v.w;
                }
            }
        } else {
            const float4* q = (const float4*)(w + (size_t)k * CDIM);
#pragma unroll
            for (int j = 0; j < 4; ++j) {
                float4 v = q[j];
                acc[4 * j + 0] = v.x;
                acc[4 * j + 1] = v.y;
                acc[4 * j + 2] = v.z;
                acc[4 * j + 3] = v.w;
            }
        }
#pragma unroll
        for (int n = 0; n < CDIM; ++n) lw[k * LDS_STRIDE + n] = acc[n];
    }
    __syncthreads();

    const int wave = tid >> 5;
    const int lane = tid & 31;
    const int tile = blockIdx.x * 8 + wave;
    if (tile >= numTiles) return;   // wave-uniform; EXEC stays all-ones for WMMA

    const int n = lane & 15;        // N / column-in-row index
    const int h = lane >> 4;        // half-wave selector (K offset +2 / M offset +8)
    const long rowBase = (long)tile * 16;

    // A fragment source: lane -> row (rowBase+n), K base 2*h; v0/v1 = consecutive K
    const float* ap = x + (rowBase + n) * F_DIM + 2 * h;
    // B fragment source: v0 = Wsum[k0+2h][n], v1 = Wsum[k0+2h+1][n]
    const float* bp = lw + (2 * h) * LDS_STRIDE + n;

    v8f c = {0.f, 0.f, 0.f, 0.f, 0.f, 0.f, 0.f, 0.f};

#pragma unroll
    for (int kc = 0; kc < F_DIM; kc += 32) {
        v2f a[8];
#pragma unroll
        for (int i = 0; i < 8; ++i)               // 8 global_load_b64 in flight
            a[i] = *(const v2f*)(ap + kc + 4 * i);
#pragma unroll
        for (int i = 0; i < 8; ++i) {
            const int k0 = kc + 4 * i;
            v2f b;
            b.x = bp[k0 * LDS_STRIDE];
            b.y = bp[k0 * LDS_STRIDE + LDS_STRIDE];
            c = __builtin_amdgcn_wmma_f32_16x16x4_f32(
                false, a[i], false, b, (short)0, c, false, false);
        }
    }

    // D layout: VGPR v, lanes 0-15 -> M=v, lanes 16-31 -> M=v+8; N=lane&15
    float* op = out + (rowBase + 8 * h) * CDIM + n;
#pragma unroll
    for (int v = 0; v < 8; ++v) op[v * CDIM] = c[v];
}

extern "C" void kernel_launch(void* const* d_in, const int* in_sizes, int n_in,
                              void* d_out, int out_size, void* d_ws, size_t ws_size,
                              hipStream_t stream) {
    const float* x    = (const float*)d_in[0];   // (B,H,W,F) fp32
    const float* caps = (const float*)d_in[1];   // (1,1,F,160) fp32
    float* out = (float*)d_out;                  // (B,H,W,16) fp32

    const int M = in_sizes[0] / F_DIM;           // 100352
    const int numTiles = (M + 15) / 16;          // 6272
    const int blocks = (numTiles + 7) / 8;       // 784

    if (ws_size >= (size_t)(F_DIM * CDIM) * sizeof(float)) {
        float* wsum = (float*)d_ws;
        caps_reduce_kernel<<<(F_DIM * CDIM + 255) / 256, 256, 0, stream>>>(caps, wsum);
        caps_gemm_kernel<false><<<blocks, 256, 0, stream>>>(x, wsum, out, numTiles);
    } else {
        caps_gemm_kernel<true><<<blocks, 256, 0, stream>>>(x, caps, out, numTiles);
    }
}